// TransformerModel_38689065402773
// MI455X (gfx1250) — compile-verified
//
#include <hip/hip_runtime.h>

// Problem constants (match reference)
#define B_  131072
#define S_  32
#define D_  16
#define WPB 4            // waves per block (wave32)
#define TPB (WPB * 32)   // 128 threads

// Padded LDS row strides (floats). Even (8B-aligned b64 fragments), rows
// 16B-aligned for b128, and non-power-of-two to spread LDS banks:
//  - XS=20: 8*20 = 160 = 32 mod 64 -> the two lane-halves of a C/D tile
//    store land in disjoint bank groups (conflict-free tile stores).
//  - SS=36: 36*l mod 64 cycles through 16 values -> softmax row walk is at
//    worst 2-way conflicted instead of 16-way at stride 32.
#define XS  20
#define SS  36

typedef float v2f __attribute__((ext_vector_type(2)));
typedef float v8f __attribute__((ext_vector_type(8)));

// D = A(16x4 f32) * B(4x16 f32) + C(16x16 f32), wave32 WMMA.
__device__ __forceinline__ v8f wmma4(v2f a, v2f b, v8f c) {
  return __builtin_amdgcn_wmma_f32_16x16x4_f32(false, a, false, b, (short)0, c,
                                               false, false);
}

// Same-wave LDS producer->consumer ordering: DS ops of one wave retire in
// order, so dscnt==0 makes prior writes visible to all lanes of this wave.
__device__ __forceinline__ void lds_fence() {
  asm volatile("s_wait_dscnt 0" ::: "memory");
}

// A-operand fragment (16x4 chunk): lane(m,half) holds A[row][col..col+1],
// contiguous in a row-major matrix -> ds_load_b64 (pairs merge to 2addr_b64).
__device__ __forceinline__ v2f frag_rowpair(const float* base, int row, int rs,
                                            int col) {
  return *(const v2f*)(base + row * rs + col);
}

// B-operand fragment when B is stored row-major *untransposed* (attn @ V):
// VGPR v holds B[row0+v][col] -> two strided ds_load_b32.
__device__ __forceinline__ v2f frag_col(const float* base, int row0, int rs,
                                        int col) {
  v2f r;
  r.x = base[(row0 + 0) * rs + col];
  r.y = base[(row0 + 1) * rs + col];
  return r;
}

// C/D 16x16 tile: VGPR v, lane(m,half) -> row (row0 + v), col.
__device__ __forceinline__ void store_tile(float* base, int rs, int row0,
                                           int col, v8f acc) {
#pragma unroll
  for (int v = 0; v < 8; ++v) base[(row0 + v) * rs + col] = acc[v];
}

__global__ __launch_bounds__(TPB) void encoder_kernel(
    const int* __restrict__ arr1, const int* __restrict__ arr2,
    const float* __restrict__ emb,
    const float* __restrict__ wq, const float* __restrict__ wk,
    const float* __restrict__ wv,
    const float* __restrict__ w1, const float* __restrict__ b1,
    const float* __restrict__ w2, const float* __restrict__ b2,
    float* __restrict__ out) {
  __shared__ __align__(16) float s_w[5][256];       // wq wk wv w1 w2
  __shared__ __align__(16) float s_b[2][16];        // b1 b2
  __shared__ __align__(16) float s_x[WPB][S_ * XS];
  __shared__ __align__(16) float s_q[WPB][S_ * XS];
  __shared__ __align__(16) float s_k[WPB][S_ * XS];
  __shared__ __align__(16) float s_v[WPB][S_ * XS];
  __shared__ __align__(16) float s_s[WPB][S_ * SS];

  const int tid = threadIdx.x;
#pragma unroll
  for (int i = tid; i < 256; i += TPB) {
    s_w[0][i] = wq[i];
    s_w[1][i] = wk[i];
    s_w[2][i] = wv[i];
    s_w[3][i] = w1[i];
    s_w[4][i] = w2[i];
  }
  if (tid < 16) {
    s_b[0][tid] = b1[tid];
    s_b[1][tid] = b2[tid];
  }
  __syncthreads();

  const int wave = tid >> 5;
  const int lane = tid & 31;
  const int m    = lane & 15;  // output column / A row (mod 16)
  const int half = lane >> 4;  // lane half selects K/row sub-group

  const unsigned task = blockIdx.x * WPB + wave;  // 0 .. 2B-1, exact
  const int enc = (task >= (unsigned)B_) ? 1 : 0;
  const int b   = enc ? (int)(task - (unsigned)B_) : (int)task;
  const int* __restrict__ arr = enc ? arr2 : arr1;

  float* X  = s_x[wave];
  float* Q  = s_q[wave];
  float* K  = s_k[wave];
  float* V  = s_v[wave];
  float* Sc = s_s[wave];

  // ---- embedding gather: lane owns sequence row `lane` (S_ == 32 lanes)
  {
    const int tok = arr[b * S_ + lane];
    const float4* er = (const float4*)(emb + (size_t)tok * D_);
    float4* xr = (float4*)(X + lane * XS);  // lane*80B, 16B aligned
    xr[0] = er[0];
    xr[1] = er[1];
    xr[2] = er[2];
    xr[3] = er[3];
  }
  lds_fence();

  // ---- Q/K/V = x @ W^T   ([32,16] = two 16x16 tiles, K=16 -> 4 wmma steps)
#pragma unroll
  for (int t = 0; t < 2; ++t) {
    v8f cq = {}, ck = {}, cv = {};
#pragma unroll
    for (int kc = 0; kc < 4; ++kc) {
      v2f a = frag_rowpair(X, 16 * t + m, XS, 4 * kc + 2 * half);
      cq = wmma4(a, frag_rowpair(s_w[0], m, D_, 4 * kc + 2 * half), cq);
      ck = wmma4(a, frag_rowpair(s_w[1], m, D_, 4 * kc + 2 * half), ck);
      cv = wmma4(a, frag_rowpair(s_w[2], m, D_, 4 * kc + 2 * half), cv);
    }
    store_tile(Q, XS, 16 * t + 8 * half, m, cq);
    store_tile(K, XS, 16 * t + 8 * half, m, ck);
    store_tile(V, XS, 16 * t + 8 * half, m, cv);
  }
  lds_fence();

  // ---- scores = Q @ K^T  ([32,32] = four 16x16 tiles, K=16)
#pragma unroll
  for (int tm = 0; tm < 2; ++tm)
#pragma unroll
    for (int tn = 0; tn < 2; ++tn) {
      v8f c = {};
#pragma unroll
      for (int kc = 0; kc < 4; ++kc) {
        v2f a  = frag_rowpair(Q, 16 * tm + m, XS, 4 * kc + 2 * half);
        v2f bb = frag_rowpair(K, 16 * tn + m, XS, 4 * kc + 2 * half);
        c = wmma4(a, bb, c);
      }
      store_tile(Sc, SS, 16 * tm + 8 * half, 16 * tn + m, c);
    }
  lds_fence();

  // ---- softmax(scores / 4): one 32-wide row per lane, held in VGPRs.
  // Row loaded/stored as 8x float4 (ds_load_b128 / ds_store_b128).
  {
    float4* rowv = (float4*)(Sc + lane * SS);  // lane*144B, 16B aligned
    float r[S_];
#pragma unroll
    for (int i = 0; i < 8; ++i) {
      float4 f = rowv[i];
      r[4 * i + 0] = f.x;
      r[4 * i + 1] = f.y;
      r[4 * i + 2] = f.z;
      r[4 * i + 3] = f.w;
    }
    float mx = -3.402823466e38f;
#pragma unroll
    for (int j = 0; j < S_; ++j) mx = fmaxf(mx, r[j]);
    float sum = 0.f;
#pragma unroll
    for (int j = 0; j < S_; ++j) {
      r[j] = __expf((r[j] - mx) * 0.25f);  // /sqrt(D)=0.25 folded in
      sum += r[j];
    }
    const float inv = 1.f / sum;
#pragma unroll
    for (int i = 0; i < 8; ++i) {
      float4 f;
      f.x = r[4 * i + 0] * inv;
      f.y = r[4 * i + 1] * inv;
      f.z = r[4 * i + 2] * inv;
      f.w = r[4 * i + 3] * inv;
      rowv[i] = f;
    }
  }
  lds_fence();

  // ---- x2 = attn @ V  (K=32 -> 8 wmma steps; V used untransposed)
#pragma unroll
  for (int t = 0; t < 2; ++t) {
    v8f c = {};
#pragma unroll
    for (int kc = 0; kc < 8; ++kc) {
      v2f a  = frag_rowpair(Sc, 16 * t + m, SS, 4 * kc + 2 * half);
      v2f bb = frag_col(V, 4 * kc + 2 * half, XS, m);
      c = wmma4(a, bb, c);
    }
    store_tile(X, XS, 16 * t + 8 * half, m, c);  // reuse X for x2
  }
  lds_fence();

  // ---- h = relu(x2 @ w1^T + b1)
  {
    const float bb1 = s_b[0][m];
#pragma unroll
    for (int t = 0; t < 2; ++t) {
      v8f c = {};
#pragma unroll
      for (int kc = 0; kc < 4; ++kc) {
        v2f a  = frag_rowpair(X, 16 * t + m, XS, 4 * kc + 2 * half);
        v2f bb = frag_rowpair(s_w[3], m, D_, 4 * kc + 2 * half);
        c = wmma4(a, bb, c);
      }
#pragma unroll
      for (int v = 0; v < 8; ++v) c[v] = fmaxf(c[v] + bb1, 0.f);
      store_tile(Q, XS, 16 * t + 8 * half, m, c);  // reuse Q for h
    }
  }
  lds_fence();

  // ---- y = h @ w2^T (+ b2), fused mean-pool over S in registers
  {
    float partial = 0.f;
#pragma unroll
    for (int t = 0; t < 2; ++t) {
      v8f c = {};
#pragma unroll
      for (int kc = 0; kc < 4; ++kc) {
        v2f a  = frag_rowpair(Q, 16 * t + m, XS, 4 * kc + 2 * half);
        v2f bb = frag_rowpair(s_w[4], m, D_, 4 * kc + 2 * half);
        c = wmma4(a, bb, c);
      }
#pragma unroll
      for (int v = 0; v < 8; ++v) partial += c[v];  // 16 of 32 rows, col m
    }
    // combine the two lane halves (other 16 rows of column m)
    float other = __shfl_xor(partial, 16, 32);
    float mean  = (partial + other) * (1.f / 32.f) + s_b[1][m];
    if (half == 0) {
      out[(size_t)enc * ((size_t)B_ * D_) + (size_t)b * D_ + m] = mean;
    }
  }
}

extern "C" void kernel_launch(void* const* d_in, const int* in_sizes, int n_in,
                              void* d_out, int out_size, void* d_ws,
                              size_t ws_size, hipStream_t stream) {
  (void)in_sizes; (void)n_in; (void)out_size; (void)d_ws; (void)ws_size;
  const int*   arr1 = (const int*)d_in[0];
  const int*   arr2 = (const int*)d_in[1];
  const float* emb  = (const float*)d_in[2];
  const float* wq   = (const float*)d_in[3];
  const float* wk   = (const float*)d_in[4];
  const float* wv   = (const float*)d_in[5];
  const float* w1   = (const float*)d_in[6];
  const float* b1   = (const float*)d_in[7];
  const float* w2   = (const float*)d_in[8];
  const float* b2   = (const float*)d_in[9];
  float* out = (float*)d_out;

  const unsigned tasks  = 2u * B_;          // 262144 waves
  const unsigned blocks = tasks / WPB;      // 65536 blocks of 128 threads
  encoder_kernel<<<dim3(blocks), dim3(TPB), 0, stream>>>(
      arr1, arr2, emb, wq, wk, wv, w1, b1, w2, b2, out);
}